// NTXent_71519795413283
// MI455X (gfx1250) — compile-verified
//
#include <hip/hip_runtime.h>
#include <hip/hip_bf16.h>
#include <math.h>

// NT-Xent (B=4096, K=512, T=0.5) for gfx1250 using V_WMMA_F32_16X16X4_F32.
//
// X = [a_hat; p_hat] scaled by sqrt(1/T) so every Gram entry is a logit and
// the diagonal is exactly 2.0. Logits bounded in [-2,2] => plain sum-of-exp
// logsumexp (no running max needed).

typedef __attribute__((ext_vector_type(2))) float v2f;
typedef __attribute__((ext_vector_type(8))) float v8f;

#define NTX_B   4096
#define NTX_N2  8192   // 2B rows of X
#define NTX_K   512
#define SQRT_INV_T 1.41421356237309504880f  // sqrt(2)
#define INV_T 2.0f

// ---------------------------------------------------------------------------
// Kernel 1: per-row L2 normalize (eps=1e-12 like torch/jax ref), scale by
// sqrt(1/T), write X = [a_hat; p_hat]; also picked[i] = <a_hat_i, p_hat_i>/T.
// One 256-thread block per row pair; each thread covers 2 of 512 elements.
// ---------------------------------------------------------------------------
__global__ __launch_bounds__(256) void ntx_normalize_kernel(
    const float* __restrict__ a, const float* __restrict__ p,
    float* __restrict__ X, float* __restrict__ picked) {
  const int row = blockIdx.x;
  const int t = threadIdx.x;
  const float* ar = a + (size_t)row * NTX_K;
  const float* pr = p + (size_t)row * NTX_K;

  float a0 = ar[t], a1 = ar[t + 256];
  float p0 = pr[t], p1 = pr[t + 256];

  float sa = a0 * a0 + a1 * a1;
  float sp = p0 * p0 + p1 * p1;
  float sd = a0 * p0 + a1 * p1;

  // wave32 butterfly reduce
  #pragma unroll
  for (int m = 1; m < 32; m <<= 1) {
    sa += __shfl_xor(sa, m, 32);
    sp += __shfl_xor(sp, m, 32);
    sd += __shfl_xor(sd, m, 32);
  }
  __shared__ float rsa[8], rsp[8], rsd[8];
  const int wave = t >> 5;
  if ((t & 31) == 0) { rsa[wave] = sa; rsp[wave] = sp; rsd[wave] = sd; }
  __syncthreads();
  float ta = 0.f, tp = 0.f, td = 0.f;
  #pragma unroll
  for (int w = 0; w < 8; ++w) { ta += rsa[w]; tp += rsp[w]; td += rsd[w]; }

  const float na = fmaxf(sqrtf(ta), 1e-12f);
  const float np = fmaxf(sqrtf(tp), 1e-12f);
  const float fa = SQRT_INV_T / na;
  const float fp = SQRT_INV_T / np;

  float* Xa = X + (size_t)row * NTX_K;
  float* Xp = X + (size_t)(row + NTX_B) * NTX_K;
  Xa[t] = a0 * fa; Xa[t + 256] = a1 * fa;
  Xp[t] = p0 * fp; Xp[t + 256] = p1 * fp;

  if (t == 0) picked[row] = (td / (na * np)) * INV_T;
}

// ---------------------------------------------------------------------------
// Kernel 2: S[i] = sum_j exp( (X X^T)[i,j] ) over all 8192 j, via f32 WMMA.
// Block = 256 threads = 8 waves; block owns 16 rows (staged in LDS, 32 KB);
// each wave sweeps 16-wide j-tiles (stride 8). Fragment layout per ISA
// 7.12.2 (32-bit A 16x4): lanes 0-15 hold K={0,1}, lanes 16-31 hold K={2,3};
// B (4x16) mirrors it with N across lanes. C: VGPR r => row M=r (lanes 0-15)
// / M=8+r (lanes 16-31).
// ---------------------------------------------------------------------------
__global__ __launch_bounds__(256) void ntx_gram_expsum_kernel(
    const float* __restrict__ X, float* __restrict__ S) {
  __shared__ float lds_a[16 * NTX_K];  // 32 KB: this block's 16 A-rows

  const int i0 = blockIdx.x * 16;
  const int tid = threadIdx.x;

  // stage A tile: 8192 floats, 256 threads * 8 float4
  {
    const float4* src = (const float4*)(X + (size_t)i0 * NTX_K);
    float4* dst = (float4*)lds_a;
    #pragma unroll
    for (int idx = tid; idx < (16 * NTX_K) / 4; idx += 256) dst[idx] = src[idx];
  }
  __syncthreads();

  const int wave = tid >> 5;     // 0..7
  const int lane = tid & 31;
  const int half = lane >> 4;    // 0: K lo pair, 1: K hi pair
  const int lr = lane & 15;      // row-within-tile / col-within-tile

  const float* aBase = lds_a + lr * NTX_K + 2 * half;
  float acc[8] = {0.f, 0.f, 0.f, 0.f, 0.f, 0.f, 0.f, 0.f};

  for (int jt = wave; jt < NTX_N2 / 16; jt += 8) {
    const int j0 = jt * 16;
    const float* bBase = X + (size_t)(j0 + lr) * NTX_K + 2 * half;

    v8f c = {0.f, 0.f, 0.f, 0.f, 0.f, 0.f, 0.f, 0.f};
    #pragma unroll 8
    for (int k = 0; k < NTX_K; k += 4) {
      v2f av = *(const v2f*)(aBase + k);   // ds_load_b64
      v2f bv = *(const v2f*)(bBase + k);   // global_load_b64 (L2-resident)
      c = __builtin_amdgcn_wmma_f32_16x16x4_f32(
          false, av, false, bv, (short)0, c, false, false);
    }
    #pragma unroll
    for (int r = 0; r < 8; ++r) acc[r] += __expf(c[r]);
  }

  // Reduce across the 16 lanes that share a row (N dimension), then atomically
  // accumulate the per-wave partial into S[row].
  #pragma unroll
  for (int r = 0; r < 8; ++r) {
    float v = acc[r];
    #pragma unroll
    for (int m = 1; m <= 8; m <<= 1) v += __shfl_xor(v, m, 32);
    if (lr == 0) atomicAdd(&S[i0 + 8 * half + r], v);
  }
}

// ---------------------------------------------------------------------------
// Kernel 3: loss = mean_i( log(S[i] - e^2) - picked[i mod B] ), single block.
// ---------------------------------------------------------------------------
__global__ __launch_bounds__(1024) void ntx_finalize_kernel(
    const float* __restrict__ S, const float* __restrict__ picked,
    float* __restrict__ out) {
  const int t = threadIdx.x;
  const float e2 = expf(2.0f);  // exp of the (dropped) diagonal logit
  double part = 0.0;
  for (int i = t; i < NTX_N2; i += 1024) {
    float s = fmaxf(S[i] - e2, 1e-30f);
    float lse = logf(s);
    part += (double)(lse - picked[i & (NTX_B - 1)]);
  }
  __shared__ double red[1024];
  red[t] = part;
  __syncthreads();
  for (int off = 512; off > 0; off >>= 1) {
    if (t < off) red[t] += red[t + off];
    __syncthreads();
  }
  if (t == 0) out[0] = (float)(red[0] / (double)NTX_N2);
}

// ---------------------------------------------------------------------------
extern "C" void kernel_launch(void* const* d_in, const int* in_sizes, int n_in,
                              void* d_out, int out_size, void* d_ws, size_t ws_size,
                              hipStream_t stream) {
  (void)in_sizes; (void)n_in; (void)out_size; (void)ws_size;
  const float* anchor   = (const float*)d_in[0];
  const float* positive = (const float*)d_in[1];
  float* out = (float*)d_out;

  // Workspace layout (floats): X[8192*512] | picked[4096] | S[8192]
  float* X      = (float*)d_ws;
  float* picked = X + (size_t)NTX_N2 * NTX_K;
  float* S      = picked + NTX_B;

  hipMemsetAsync(S, 0, NTX_N2 * sizeof(float), stream);

  ntx_normalize_kernel<<<NTX_B, 256, 0, stream>>>(anchor, positive, X, picked);
  ntx_gram_expsum_kernel<<<NTX_N2 / 16, 256, 0, stream>>>(X, S);
  ntx_finalize_kernel<<<1, 1024, 0, stream>>>(S, picked, out);
}